// ButterflyRotation_68994354643574
// MI455X (gfx1250) — compile-verified
//
#include <hip/hip_runtime.h>
#include <math.h>
#include <stdint.h>

#define DIM        1024
#define STAGES     10
#define NANG       (STAGES * (DIM / 2))   // 5120 angles total
#define CS_BYTES   (NANG * 8)             // 40960 B cos/sin float2 table
#define P1ANG      2560                   // angles belonging to stages 0..4 (permuted region)
#define GRPB       144                    // padded bytes per 32-float group (36 floats)
#define INBUF      (32 * GRPB)            // 4608 B per row input buffer
#define WBUF       (2 * INBUF)            // 9216 B per-wave buffer (A+B, overlaid by transpose)
#define TRS        272                    // byte stride in transpose buf (32*272=8704<=9216, 16B aligned)
#define NWAVES_BLK 8
#define BLK        256
#define CS_ITERS   (NANG / BLK)           // 20, exact
#define SMEM_TOTAL (CS_BYTES + NWAVES_BLK * WBUF) // 114688 B

// Async copy of one 4KB row (coalesced 512B/instr on the global side) into padded LDS:
// element e lands at lds_base + (e>>5)*144 + (e&31)*4 (4-dword chunks never cross a pad).
__device__ __forceinline__ void issue_async_row(const float* __restrict__ grow,
                                                uint32_t lds_base, int lane) {
  uint32_t laddr = lds_base + (uint32_t)(((lane >> 3) * GRPB) + ((lane & 7) * 16));
  uint32_t goff  = (uint32_t)(lane * 16);
#pragma unroll
  for (int m = 0; m < 8; ++m) {
    asm volatile("global_load_async_to_lds_b128 %0, %1, %2"
                 :: "v"(laddr), "v"(goff), "s"(grow) : "memory");
    laddr += 576u;   // 4*144: LDS advances 4 padded groups per 512B of global
    goff  += 512u;
  }
}

__global__ void __launch_bounds__(BLK) butterfly_kernel(const float* __restrict__ x,
                                                        const float* __restrict__ angles,
                                                        float* __restrict__ out,
                                                        int batch) {
  extern __shared__ char smem[];
  float2* cs = (float2*)smem;
  const int tid = threadIdx.x;

  // One-time cos/sin table, CONTROL-UNIFORM loop (k is a scalar compile-time-bounded
  // counter; no divergent loop structurization). Stages 0..4 stored permuted so
  // kernel-side angle loads are lane-contiguous (bank-conflict-free):
  //   pass1: angle flat idx  s*512 + lane*16 + p  ->  cs[(s*16+p)*32 + lane]
  //   pass2: identity layout (already lane-fastest: idx = p*32 + lane)
  for (int k = 0; k < CS_ITERS; ++k) {
    const int t = tid + k * BLK;               // < NANG for all tid < BLK
    const float ang = angles[t];
    const float2 v = make_float2(__builtin_cosf(ang), __builtin_sinf(ang));
    const int s    = t >> 9;
    const int idx  = t & 511;
    const int perm = ((s * 16 + (idx & 15)) << 5) | (idx >> 4);
    const int dst  = (t < P1ANG) ? perm : t;   // compile-time decidable per unrolled k
    cs[dst] = v;
  }
  __syncthreads();

  const int lane = tid & 31;
  const int w    = tid >> 5;
  char* wbuf = smem + CS_BYTES + w * WBUF;
  const char* inA = wbuf;
  const char* inB = wbuf + INBUF;
  char* trb = wbuf; // transpose buffer overlays input buffers (lifetimes disjoint)
  const uint32_t lds_u32 = (uint32_t)(CS_BYTES + w * WBUF);

  const float2* cs1L = cs + lane;          // pass1: cs1L[(s*16+p)*32], 256B contiguous/wave
  const float2* cs2L = cs + P1ANG + lane;  // pass2: cs2L[(s-5)*512 + p*32], contiguous/wave

  const int nunits = batch >> 1;
  const int nwaves = gridDim.x * NWAVES_BLK;
  int unit = blockIdx.x * NWAVES_BLK + w;

  if (unit < nunits) {
    const float* rowA = x + (size_t)(2 * unit) * DIM;
    issue_async_row(rowA,       lds_u32,         lane);
    issue_async_row(rowA + DIM, lds_u32 + INBUF, lane);
  }

  for (; unit < nunits; unit += nwaves) {
    asm volatile("s_wait_asynccnt 0x0" ::: "memory");

    // Pass-1 layout: lane holds contiguous elements [32*lane, 32*lane+32) of rows A,B.
    float a[32], b[32];
#pragma unroll
    for (int j = 0; j < 8; ++j) {
      float4 ta = *(const float4*)(inA + lane * GRPB + j * 16);
      float4 tb = *(const float4*)(inB + lane * GRPB + j * 16);
      a[4*j+0]=ta.x; a[4*j+1]=ta.y; a[4*j+2]=ta.z; a[4*j+3]=ta.w;
      b[4*j+0]=tb.x; b[4*j+1]=tb.y; b[4*j+2]=tb.z; b[4*j+3]=tb.w;
    }

    // Stages 0..4: partner = reg ^ (1<<s); angle = cs1[(s*16+p)*32 + lane].
#pragma unroll
    for (int s = 0; s < 5; ++s) {
#pragma unroll
      for (int p = 0; p < 16; ++p) {
        const int lowmask = (1 << s) - 1;
        const int r0 = ((p >> s) << (s + 1)) | (p & lowmask);
        const int r1 = r0 | (1 << s);
        const float2 t = cs1L[(s * 16 + p) * 32];
        const float c = t.x, si = t.y;
        const float ai = a[r0], aj = a[r1], bi = b[r0], bj = b[r1];
        a[r0] = fmaf(c,  ai, -(si * aj));
        a[r1] = fmaf(si, ai,   c * aj);
        b[r0] = fmaf(c,  bi, -(si * bj));
        b[r1] = fmaf(si, bi,   c * bj);
      }
    }

    // Transpose: element i=(32*lane+r) of (A,B) as float2 at trb + r*272 + lane*8
    // (write: banks 2L..2L+1 -> conflict-free; read b128: exactly 2-cycle optimal).
#pragma unroll
    for (int r = 0; r < 32; ++r) {
      *(float2*)(trb + r * TRS + lane * 8) = make_float2(a[r], b[r]);
    }
    asm volatile("" ::: "memory"); // order aliasing LDS store->load across TBAA types

    // Pass-2 layout: reg k holds element k*32+lane of both rows (float2).
    float2 v[32];
#pragma unroll
    for (int k = 0; k < 32; k += 2) {
      float4 t = *(const float4*)(trb + lane * TRS + k * 8);
      v[k]     = make_float2(t.x, t.y);
      v[k + 1] = make_float2(t.z, t.w);
    }

    // Buffer dead once the reads above retire: drain DS, then prefetch next unit.
    asm volatile("s_wait_dscnt 0x0" ::: "memory");
    {
      const int nxt = unit + nwaves;
      if (nxt < nunits) {
        const float* rowA = x + (size_t)(2 * nxt) * DIM;
        issue_async_row(rowA,       lds_u32,         lane);
        issue_async_row(rowA + DIM, lds_u32 + INBUF, lane);
      }
    }

    // Stages 5..9: partner = reg ^ (1<<(s-5)); angle = cs[s*512 + p*32 + lane].
#pragma unroll
    for (int s = 5; s < 10; ++s) {
      const int sb = s - 5;
#pragma unroll
      for (int p = 0; p < 16; ++p) {
        const int lowmask = (1 << sb) - 1;
        const int k0 = ((p >> sb) << (sb + 1)) | (p & lowmask);
        const int k1 = k0 | (1 << sb);
        const float2 t = cs2L[sb * 512 + p * 32];
        const float c = t.x, si = t.y;
        const float2 vi = v[k0], vj = v[k1];
        v[k0].x = fmaf(c,  vi.x, -(si * vj.x));
        v[k0].y = fmaf(c,  vi.y, -(si * vj.y));
        v[k1].x = fmaf(si, vi.x,   c * vj.x);
        v[k1].y = fmaf(si, vi.y,   c * vj.y);
      }
    }

    // Coalesced non-temporal stores (element k*32+lane).
    float* outA = out + (size_t)(2 * unit) * DIM;
    float* outB = outA + DIM;
#pragma unroll
    for (int k = 0; k < 32; ++k) {
      __builtin_nontemporal_store(v[k].x, &outA[k * 32 + lane]);
      __builtin_nontemporal_store(v[k].y, &outB[k * 32 + lane]);
    }
  }
}

extern "C" void kernel_launch(void* const* d_in, const int* in_sizes, int n_in,
                              void* d_out, int out_size, void* d_ws, size_t ws_size,
                              hipStream_t stream) {
  (void)n_in; (void)d_ws; (void)ws_size; (void)out_size;
  const float* x      = (const float*)d_in[0];
  const float* angles = (const float*)d_in[1];
  float* out          = (float*)d_out;
  const int batch = in_sizes[0] / DIM;

  // Allow >64KB dynamic LDS (112 KB). Idempotent; safe under graph capture.
  (void)hipFuncSetAttribute((const void*)butterfly_kernel,
                            hipFuncAttributeMaxDynamicSharedMemorySize, SMEM_TOTAL);

  const int nunits = batch >> 1;
  int blocks = (nunits + NWAVES_BLK - 1) / NWAVES_BLK;
  if (blocks > 512) blocks = 512;
  if (blocks < 1) blocks = 1;

  butterfly_kernel<<<dim3(blocks), dim3(BLK), SMEM_TOTAL, stream>>>(x, angles, out, batch);
}